// E3PhisNetInteraction_29609504538685
// MI455X (gfx1250) — compile-verified
//
#include <hip/hip_runtime.h>

typedef float v2f __attribute__((ext_vector_type(2)));
typedef float v8f __attribute__((ext_vector_type(8)));

#define N_NODES  20000
#define N_EDGES  256000
#define NCH      32
#define FDIM     32
#define NA       96      // C*3 columns (wa)
#define NB       352     // C*11 columns (wb)
#define LDSW     452     // 448 padded so M and M+8 rows hit disjoint LDS banks
#define TILE_E   16
#define NT_A     6
#define NT       28      // 6 + 22 N-tiles of 16

// ---- real-basis CG contractions (values = sqrt(2*l3+1) * real-w3j tensor) ----
__device__ __forceinline__ void tp121(const float* x, const float* y, float* o) {
  const float T1 = 0.316227766016838f, T2 = 0.632455532033676f, T3 = 0.547722557505166f;
  o[0] = -T1*x[0]*y[2] + T3*(x[1]*y[1] + x[2]*y[0] - x[0]*y[4]);
  o[1] =  T2*x[1]*y[2] + T3*(x[2]*y[3] + x[0]*y[1]);
  o[2] = -T1*x[2]*y[2] + T3*(x[1]*y[3] + x[2]*y[4] + x[0]*y[0]);
}
__device__ __forceinline__ void tp211(const float* x, const float* y, float* o) {
  const float T1 = 0.316227766016838f, T2 = 0.632455532033676f, T3 = 0.547722557505166f;
  o[0] = -T1*x[2]*y[0] + T3*(x[1]*y[1] + x[0]*y[2] - x[4]*y[0]);
  o[1] =  T2*x[2]*y[1] + T3*(x[3]*y[2] + x[1]*y[0]);
  o[2] = -T1*x[2]*y[2] + T3*(x[3]*y[1] + x[4]*y[2] + x[0]*y[0]);
}
__device__ __forceinline__ void tp112(const float* x, const float* y, float* o) {
  const float U1 = 0.408248290463863f, U2 = 0.816496580927726f, U3 = 0.707106781186548f;
  o[0] =  U3*(x[0]*y[2] + x[2]*y[0]);
  o[1] =  U3*(x[0]*y[1] + x[1]*y[0]);
  o[2] = -U1*(x[0]*y[0] + x[2]*y[2]) + U2*x[1]*y[1];
  o[3] =  U3*(x[1]*y[2] + x[2]*y[1]);
  o[4] =  U3*(x[2]*y[2] - x[0]*y[0]);
}
__device__ __forceinline__ void tp222(const float* x, const float* y, float* o) {
  const float V1 = 0.267261241912424f, V2 = 0.534522483824849f, V3 = 0.462910049886276f;
  o[0] =  V2*(x[0]*y[2] + x[2]*y[0]) - V3*(x[3]*y[1] + x[1]*y[3]);
  o[1] = -V1*(x[1]*y[2] + x[2]*y[1]) + V3*(x[4]*y[1] + x[1]*y[4]) - V3*(x[0]*y[3] + x[3]*y[0]);
  o[2] =  V2*(x[0]*y[0] + x[4]*y[4] - x[2]*y[2]) - V1*(x[1]*y[1] + x[3]*y[3]);
  o[3] = -V1*(x[3]*y[2] + x[2]*y[3]) - V3*(x[4]*y[3] + x[3]*y[4]) - V3*(x[0]*y[1] + x[1]*y[0]);
  o[4] =  V2*(x[4]*y[2] + x[2]*y[4]) + V3*(x[1]*y[1] - x[3]*y[3]);
}

// selfmix for a channel-dim-1 input (Y_edge): 9 -> 9
__device__ __forceinline__ void selfmix9(float y0, const float* y1, const float* y2,
    const float* w0, const float* w1, const float* w2, float b0, const float* k, float* z) {
  const float RS3 = 0.577350269189626f;  // 1/sqrt(3)
  const float S5  = 0.447213595499958f;  // 1/sqrt(5)
  float d1 = y1[0]*y1[0] + y1[1]*y1[1] + y1[2]*y1[2];
  float d2 = y2[0]*y2[0] + y2[1]*y2[1] + y2[2]*y2[2] + y2[3]*y2[3] + y2[4]*y2[4];
  z[0] = (w0[0]*y0*y0 - w0[1]*RS3*d1 + w0[2]*S5*d2) * RS3 + b0 + k[0]*y0;
  float t12[3], t21[3], t11[5], t22[5];
  tp121(y1, y2, t12);
  tp211(y2, y1, t21);
  tp112(y1, y1, t11);
  tp222(y2, y2, t22);
  float w01 = -(w1[0] + w1[1]);
#pragma unroll
  for (int j = 0; j < 3; ++j)
    z[1+j] = (w01*y0*y1[j] + w1[2]*t12[j] + w1[3]*t21[j]) * 0.5f + k[1]*y1[j];
  float w02 = (w2[0] + w2[2]);
#pragma unroll
  for (int j = 0; j < 5; ++j)
    z[4+j] = (w02*y0*y2[j] + w2[1]*t11[j] + w2[3]*t22[j]) * 0.5f + k[2]*y2[j];
}

__global__ void init_out_kernel(const float4* __restrict__ src, float4* __restrict__ dst, int n) {
  int i = blockIdx.x * blockDim.x + threadIdx.x;
  if (i < n) dst[i] = src[i];
}

__global__ __launch_bounds__(512) void e3_fused_kernel(
    const float* __restrict__ nodes_snd,
    const int*   __restrict__ edge_ind,
    const float* __restrict__ Y_edge,
    const float* __restrict__ dist_feat,
    const float* __restrict__ sm_a_w0, const float* __restrict__ sm_a_w1,
    const float* __restrict__ sm_a_w2, const float* __restrict__ sm_a_b0,
    const float* __restrict__ sm_a_k,  const float* __restrict__ cm_a_W,
    const float* __restrict__ cm_a_b,
    const float* __restrict__ sm_b_w0, const float* __restrict__ sm_b_w1,
    const float* __restrict__ sm_b_w2, const float* __restrict__ sm_b_b0,
    const float* __restrict__ sm_b_k,  const float* __restrict__ cm_b_W,
    const float* __restrict__ cm_b_b,
    const float* __restrict__ Wd_a, const float* __restrict__ bd_a,
    const float* __restrict__ Wd_b, const float* __restrict__ bd_b,
    float* __restrict__ out)
{
  __shared__ float ldsW[TILE_E * LDSW];   // 16 x (96|352) GEMM results, padded
  __shared__ float ldsZ[TILE_E][18];      // za[9], zb[9] per edge
  __shared__ int   ldsSrc[TILE_E];
  __shared__ int   ldsRecv[TILE_E];

  const int tid    = threadIdx.x;
  const int e_base = blockIdx.x * TILE_E;

  // ---------------- Stage 1: WMMA GEMM into LDS ----------------
  const int lane = tid & 31;
  const int wave = tid >> 5;
  const int row  = lane & 15;   // M (edge in tile) for A, N (column) for B/C/D
  const int kh   = lane >> 4;   // K-pair selector (A/B), +8 row offset (C/D)

  // A fragments: dist_feat tile, 16 rows x K=32, 8 chunks of K=4
  v2f afrag[8];
  {
    const float* dfp = dist_feat + (size_t)(e_base + row) * FDIM + kh * 2;
#pragma unroll
    for (int k4 = 0; k4 < 8; ++k4)
      afrag[k4] = *(const v2f*)(dfp + k4 * 4);
  }

  for (int t = wave; t < NT; t += 16) {
    const bool  isA   = (t < NT_A);
    const float* W    = isA ? Wd_a : Wd_b;
    const float* bb   = isA ? bd_a : bd_b;
    const int   ncols = isA ? NA : NB;
    const int   nb    = isA ? t * 16 : (t - NT_A) * 16;
    const int   ncol  = nb + row;

    float bias = bb[ncol];
    v8f acc;
#pragma unroll
    for (int v = 0; v < 8; ++v) acc[v] = bias;

    const float* wp = W + (size_t)(kh * 2) * ncols + ncol;
#pragma unroll
    for (int k4 = 0; k4 < 8; ++k4) {
      v2f bfrag;
      bfrag.x = wp[(k4 * 4 + 0) * ncols];
      bfrag.y = wp[(k4 * 4 + 1) * ncols];
      acc = __builtin_amdgcn_wmma_f32_16x16x4_f32(
          false, afrag[k4], false, bfrag, (short)0, acc, false, false);
    }

    const int ldcol = (isA ? 0 : NA) + ncol;
#pragma unroll
    for (int v = 0; v < 8; ++v) {
      int m = v + kh * 8;
      ldsW[m * LDSW + ldcol] = acc[v];
    }
  }

  // ---------------- Stage 1.5: per-edge z vectors + edge indices ----------------
  if (tid < TILE_E) {
    int e = e_base + tid;
    ldsSrc[tid]  = edge_ind[2 * e + 0];
    ldsRecv[tid] = edge_ind[2 * e + 1];
    const float* y = Y_edge + (size_t)e * 9;
    float y0 = y[0];
    float y1[3] = { y[1], y[2], y[3] };
    float y2[5] = { y[4], y[5], y[6], y[7], y[8] };
    selfmix9(y0, y1, y2, sm_a_w0, sm_a_w1, sm_a_w2, sm_a_b0[0], sm_a_k, &ldsZ[tid][0]);
    selfmix9(y0, y1, y2, sm_b_w0, sm_b_w1, sm_b_w2, sm_b_b0[0], sm_b_k, &ldsZ[tid][9]);
  }

  __syncthreads();

  // ---------------- Stage 2: per (edge, channel) CG math + scatter ----------------
  {
    const int e = tid >> 5;      // 0..15
    const int c = tid & 31;      // channel
    const int src  = ldsSrc[e];
    const int recv = ldsRecv[e];

    float n[9];
    {
      const float* np = nodes_snd + ((size_t)src * NCH + c) * 9;
#pragma unroll
      for (int m = 0; m < 9; ++m) n[m] = np[m];
    }
    const float* za = &ldsZ[e][0];
    const float* zb = &ldsZ[e][9];

    float Ya[9], Yb[9];
    {
      float w0 = cm_a_W[c], w1 = cm_a_W[32 + c], w2 = cm_a_W[64 + c];
      Ya[0] = za[0] * w0 + cm_a_b[c];
#pragma unroll
      for (int j = 0; j < 3; ++j) Ya[1+j] = za[1+j] * w1;
#pragma unroll
      for (int j = 0; j < 5; ++j) Ya[4+j] = za[4+j] * w2;
    }
    {
      float w0 = cm_b_W[c], w1 = cm_b_W[32 + c], w2 = cm_b_W[64 + c];
      Yb[0] = zb[0] * w0 + cm_b_b[c];
#pragma unroll
      for (int j = 0; j < 3; ++j) Yb[1+j] = zb[1+j] * w1;
#pragma unroll
      for (int j = 0; j < 5; ++j) Yb[4+j] = zb[4+j] * w2;
    }

    const float* wrow = &ldsW[e * LDSW];

    // a-branch: wa * n0 gating
    float wl0 = wrow[c*3 + 0] * n[0];
    float wl1 = wrow[c*3 + 1] * n[0];
    float wl2 = wrow[c*3 + 2] * n[0];
    float ov[9];
    ov[0] = wl0 * Ya[0];
#pragma unroll
    for (int j = 0; j < 3; ++j) ov[1+j] = wl1 * Ya[1+j];
#pragma unroll
    for (int j = 0; j < 5; ++j) ov[4+j] = wl2 * Ya[4+j];

    // b-branch: pair_mix(neighbors, Yb, wb)
    float wb[11];
#pragma unroll
    for (int i = 0; i < 11; ++i) wb[i] = wrow[NA + c*11 + i];

    const float RS3 = 0.577350269189626f;
    const float S5  = 0.447213595499958f;
    float d1 = n[1]*Yb[1] + n[2]*Yb[2] + n[3]*Yb[3];
    float d2 = n[4]*Yb[4] + n[5]*Yb[5] + n[6]*Yb[6] + n[7]*Yb[7] + n[8]*Yb[8];
    ov[0] += (wb[0]*n[0]*Yb[0] - wb[1]*RS3*d1 + wb[2]*S5*d2) * RS3;

    float t12[3], t21[3], t11[5], t22[5];
    tp121(&n[1], &Yb[4], t12);
    tp211(&n[4], &Yb[1], t21);
    tp112(&n[1], &Yb[1], t11);
    tp222(&n[4], &Yb[4], t22);
#pragma unroll
    for (int j = 0; j < 3; ++j)
      ov[1+j] += (-wb[3]*n[0]*Yb[1+j] - wb[4]*n[1+j]*Yb[0]
                  + wb[5]*t12[j] + wb[6]*t21[j]) * 0.5f;
#pragma unroll
    for (int j = 0; j < 5; ++j)
      ov[4+j] += (wb[7]*n[0]*Yb[4+j] + wb[8]*t11[j]
                  + wb[9]*n[4+j]*Yb[0] + wb[10]*t22[j]) * 0.5f;

    float* op = out + ((size_t)recv * NCH + c) * 9;
#pragma unroll
    for (int m = 0; m < 9; ++m) atomicAdd(op + m, ov[m]);
  }
}

extern "C" void kernel_launch(void* const* d_in, const int* in_sizes, int n_in,
                              void* d_out, int out_size, void* d_ws, size_t ws_size,
                              hipStream_t stream) {
  (void)in_sizes; (void)n_in; (void)out_size; (void)d_ws; (void)ws_size;
  const float* nodes_rec = (const float*)d_in[0];
  const float* nodes_snd = (const float*)d_in[1];
  const int*   edge_ind  = (const int*)  d_in[2];
  const float* Y_edge    = (const float*)d_in[3];
  const float* dist_feat = (const float*)d_in[4];
  const float* sm_a_w0 = (const float*)d_in[5];
  const float* sm_a_w1 = (const float*)d_in[6];
  const float* sm_a_w2 = (const float*)d_in[7];
  const float* sm_a_b0 = (const float*)d_in[8];
  const float* sm_a_k  = (const float*)d_in[9];
  const float* cm_a_W  = (const float*)d_in[10];
  const float* cm_a_b  = (const float*)d_in[11];
  const float* sm_b_w0 = (const float*)d_in[12];
  const float* sm_b_w1 = (const float*)d_in[13];
  const float* sm_b_w2 = (const float*)d_in[14];
  const float* sm_b_b0 = (const float*)d_in[15];
  const float* sm_b_k  = (const float*)d_in[16];
  const float* cm_b_W  = (const float*)d_in[17];
  const float* cm_b_b  = (const float*)d_in[18];
  const float* Wd_a    = (const float*)d_in[19];
  const float* bd_a    = (const float*)d_in[20];
  const float* Wd_b    = (const float*)d_in[21];
  const float* bd_b    = (const float*)d_in[22];
  float* out = (float*)d_out;

  // out := nodes_rec (harness poisons d_out; we accumulate with atomics)
  int n4 = (N_NODES * NCH * 9) / 4;
  init_out_kernel<<<(n4 + 255) / 256, 256, 0, stream>>>(
      (const float4*)nodes_rec, (float4*)out, n4);

  dim3 grid(N_EDGES / TILE_E);   // 16000 blocks, 512 threads (16 waves) each
  e3_fused_kernel<<<grid, 512, 0, stream>>>(
      nodes_snd, edge_ind, Y_edge, dist_feat,
      sm_a_w0, sm_a_w1, sm_a_w2, sm_a_b0, sm_a_k, cm_a_W, cm_a_b,
      sm_b_w0, sm_b_w1, sm_b_w2, sm_b_b0, sm_b_k, cm_b_W, cm_b_b,
      Wd_a, bd_a, Wd_b, bd_b, out);
}